// TreeCRF_15255723835571
// MI455X (gfx1250) — compile-verified
//
#include <hip/hip_runtime.h>
#include <hip/hip_bf16.h>
#include <math.h>

// Problem constants (from reference)
#define N_LEAVES 16384
#define N_NODES  (2 * N_LEAVES - 1)   // 32767
#define DIM      1024
#define NLAB     32

typedef __attribute__((ext_vector_type(2))) float v2f;
typedef __attribute__((ext_vector_type(8))) float v8f;
typedef __attribute__((ext_vector_type(4))) int   v4i;

typedef __attribute__((address_space(1))) v4i as1_v4i;
typedef __attribute__((address_space(3))) v4i as3_v4i;

#if defined(__has_builtin)
#if __has_builtin(__builtin_amdgcn_global_load_async_to_lds_b128) && \
    __has_builtin(__builtin_amdgcn_s_wait_asynccnt)
#define HAVE_ASYNC_LDS 1
#endif
#endif

// ---------------------------------------------------------------------------
// Kernel 1: fused emission GEMM (V_WMMA_F32_16X16X4_F32) + per-block column
// partial sums of hidden, with double-buffered async global->LDS staging.
//   scores_raw[n][l] = sum_k hidden[n][k] * W_pred[l][k]        (no const yet)
//   partial[blk][d]  = sum over this block's 64 rows of hidden[row][d]
// block = 128 threads (4 waves); wave w owns rows [wg*64 + w*16, +16).
// Chunk = 64 K-columns. 12 async B128 ops per thread per chunk; the next
// chunk is prefetched while the current one is consumed (wait asynccnt<=12).
// ---------------------------------------------------------------------------
#define KC 64
#define LSTRIDE 68
#define NCHUNK (DIM / KC)                   // 16
#define ASYNC_PER_CHUNK 12                  // 8 hidden + 4 W per thread

#if defined(HAVE_ASYNC_LDS)
__device__ __forceinline__ void stage_async(const float* __restrict__ hidden,
                                            const float* __restrict__ W_pred,
                                            float* bh, float* bw,
                                            int wg_row0, int kbase, int tid) {
    #pragma unroll
    for (int i = 0; i < 8; ++i) {
        int idx = i * 128 + tid;
        int r   = idx >> 4;                 // 0..63
        int c4  = idx & 15;                 // float4 units
        int gr  = wg_row0 + r;
        if (gr > N_NODES - 1) gr = N_NODES - 1;     // clamp (fixed later)
        __builtin_amdgcn_global_load_async_to_lds_b128(
            (as1_v4i*)(hidden + (size_t)gr * DIM + kbase + c4 * 4),
            (as3_v4i*)(bh + r * LSTRIDE + c4 * 4), 0, 0);
    }
    #pragma unroll
    for (int i = 0; i < 4; ++i) {
        int idx = i * 128 + tid;
        int r   = idx >> 4;                 // 0..31
        int c4  = idx & 15;
        __builtin_amdgcn_global_load_async_to_lds_b128(
            (as1_v4i*)(W_pred + (size_t)r * (2 * DIM) + kbase + c4 * 4),
            (as3_v4i*)(bw + r * LSTRIDE + c4 * 4), 0, 0);
    }
}
#endif

__global__ void emission_wmma_kernel(const float* __restrict__ hidden,
                                     const float* __restrict__ W_pred,
                                     float* __restrict__ scores,
                                     float* __restrict__ partial) {
#if defined(HAVE_ASYNC_LDS)
    __shared__ float lds_h[2][64 * LSTRIDE];    // 2 x 17408 B
    __shared__ float lds_w[2][32 * LSTRIDE];    // 2 x  8704 B
#else
    __shared__ float lds_h[1][64 * LSTRIDE];
    __shared__ float lds_w[1][32 * LSTRIDE];
#endif
    __shared__ float cs_buf[KC];                //    256 B

    const int tid  = threadIdx.x;           // 0..127
    const int wave = tid >> 5;              // 0..3
    const int lane = tid & 31;

    const int wg_row0   = blockIdx.x * 64;
    const int wave_row0 = wg_row0 + wave * 16;

    v8f acc0 = {0.f,0.f,0.f,0.f,0.f,0.f,0.f,0.f};   // N = 0..15
    v8f acc1 = {0.f,0.f,0.f,0.f,0.f,0.f,0.f,0.f};   // N = 16..31

    const int rl  = lane & 15;              // row within M-tile / col within N-tile
    const int khi = (lane >> 4) << 1;       // lanes 16..31 carry K+2,K+3

    // colsum assignment: thread owns (column c of chunk, row-half h)
    const int cc = tid & 63;                // 0..63
    const int hh = tid >> 6;                // 0..1

#if defined(HAVE_ASYNC_LDS)
    stage_async(hidden, W_pred, lds_h[0], lds_w[0], wg_row0, 0, tid);
#endif

    for (int chunk = 0; chunk < NCHUNK; ++chunk) {
        const int kbase = chunk * KC;
#if defined(HAVE_ASYNC_LDS)
        const int cur = chunk & 1;
        const int nxt = cur ^ 1;
        if (chunk < NCHUNK - 1) {
            stage_async(hidden, W_pred, lds_h[nxt], lds_w[nxt],
                        wg_row0, kbase + KC, tid);
            __builtin_amdgcn_s_wait_asynccnt(ASYNC_PER_CHUNK);  // cur chunk done
        } else {
            __builtin_amdgcn_s_wait_asynccnt(0);
        }
        const float* bh = lds_h[cur];
        const float* bw = lds_w[cur];
#else
        const int cur = 0;
        // fallback: batch all loads into registers, then store (keeps MLP)
        float4 hv[8];
        #pragma unroll
        for (int i = 0; i < 8; ++i) {
            int idx = i * 128 + tid;
            int r   = idx >> 4, c4 = idx & 15;
            int gr  = wg_row0 + r;
            if (gr > N_NODES - 1) gr = N_NODES - 1;
            hv[i] = *(const float4*)&hidden[(size_t)gr * DIM + kbase + c4 * 4];
        }
        float4 wv[4];
        #pragma unroll
        for (int i = 0; i < 4; ++i) {
            int idx = i * 128 + tid;
            int r   = idx >> 4, c4 = idx & 15;
            wv[i] = *(const float4*)&W_pred[(size_t)r * (2 * DIM) + kbase + c4 * 4];
        }
        #pragma unroll
        for (int i = 0; i < 8; ++i) {
            int idx = i * 128 + tid;
            int r   = idx >> 4, c4 = idx & 15;
            *(float4*)&lds_h[0][r * LSTRIDE + c4 * 4] = hv[i];
        }
        #pragma unroll
        for (int i = 0; i < 4; ++i) {
            int idx = i * 128 + tid;
            int r   = idx >> 4, c4 = idx & 15;
            *(float4*)&lds_w[0][r * LSTRIDE + c4 * 4] = wv[i];
        }
        const float* bh = lds_h[0];
        const float* bw = lds_w[0];
#endif
        __syncthreads();                    // B1: current chunk resident in LDS

        // --- 16 WMMA k-steps of K=4 over this chunk ---
        #pragma unroll
        for (int ks = 0; ks < KC; ks += 4) {
            const int kk = ks + khi;
            const float* ha = &bh[(wave * 16 + rl) * LSTRIDE + kk];
            v2f a;  a.x = ha[0];  a.y = ha[1];
            const float* w0 = &bw[rl * LSTRIDE + kk];
            v2f b0; b0.x = w0[0]; b0.y = w0[1];
            const float* w1 = &bw[(16 + rl) * LSTRIDE + kk];
            v2f b1; b1.x = w1[0]; b1.y = w1[1];
            acc0 = __builtin_amdgcn_wmma_f32_16x16x4_f32(false, a, false, b0,
                                                         (short)0, acc0, false, false);
            acc1 = __builtin_amdgcn_wmma_f32_16x16x4_f32(false, a, false, b1,
                                                         (short)0, acc1, false, false);
        }

        // --- column partial sums of this chunk (for avg_h) ---
        float csum = 0.0f;
        #pragma unroll
        for (int r = 0; r < 32; ++r)
            csum += bh[(hh * 32 + r) * LSTRIDE + cc];
        if (hh == 1) cs_buf[cc] = csum;
        __syncthreads();                    // B2: cs_buf ready; buffer reusable
        if (hh == 0)
            partial[(size_t)blockIdx.x * DIM + kbase + cc] = csum + cs_buf[cc];
        (void)cur;
    }

    // store raw emission (const32 folded in later by the DP kernels).
    #pragma unroll
    for (int r = 0; r < 8; ++r) {
        const int m   = r + ((lane >> 4) << 3);   // C layout: lanes 16..31 -> M+8
        const int row = wave_row0 + m;
        if (row < N_NODES) {
            scores[(size_t)row * NLAB + rl]      = acc0[r];
            scores[(size_t)row * NLAB + 16 + rl] = acc1[r];
        }
    }
}

// ---------------------------------------------------------------------------
// Kernel 2: reduce per-block partials -> colsum, correcting for the clamped
// duplicate pad row (row N_NODES-1 counted twice by the last block).
// grid = 32 blocks (32 cols each), block = 1024 threads. Coalesced reads.
// ---------------------------------------------------------------------------
__global__ void colsum_reduce_kernel(const float* __restrict__ partial,
                                     const float* __restrict__ hidden,
                                     float* __restrict__ colsum) {
    __shared__ float red[1024];
    const int t     = threadIdx.x;
    const int c     = blockIdx.x * 32 + (t & 31);
    const int slice = t >> 5;               // 0..31
    float acc = 0.0f;
    for (int b = slice; b < 512; b += 32)
        acc += partial[(size_t)b * DIM + c];
    red[t] = acc;
    __syncthreads();
    if (t < 32) {
        float s = 0.0f;
        #pragma unroll
        for (int k = 0; k < 32; ++k) s += red[k * 32 + t];
        const int col = blockIdx.x * 32 + t;
        colsum[col] = s - hidden[(size_t)(N_NODES - 1) * DIM + col];
    }
}

// ---------------------------------------------------------------------------
// Kernel 3: const32[l] = b_pred[l] + (1/N_NODES) * sum_d colsum[d]*W_pred[l][DIM+d]
// one block of 1024 threads.
// ---------------------------------------------------------------------------
__global__ void const_vec_kernel(const float* __restrict__ colsum,
                                 const float* __restrict__ W_pred,
                                 const float* __restrict__ b_pred,
                                 float* __restrict__ const32) {
    __shared__ float cs[DIM];
    const int t = threadIdx.x;
    cs[t] = colsum[t];
    __syncthreads();
    if (t < NLAB) {
        const float* w2 = &W_pred[(size_t)t * (2 * DIM) + DIM];
        float dot = 0.0f;
        for (int d = 0; d < DIM; ++d) dot += cs[d] * w2[d];
        const32[t] = b_pred[t] + dot * (1.0f / (float)N_NODES);
    }
}

// ---------------------------------------------------------------------------
// Kernel 4: one tree DP level. One wave per internal node.
// Stored child values: leaves hold RAW emission (add const32), internal
// children already hold final scores. This node's final score =
//   raw[node] + const32[l] + lse(trans[l,:]+ls) + lse(trans[l,:]+rs)
// ---------------------------------------------------------------------------
__global__ void level_kernel(float* __restrict__ scores,
                             const float* __restrict__ trans,
                             const float* __restrict__ const32,
                             const int* __restrict__ left_idx,
                             const int* __restrict__ right_idx,
                             int pos, int node_base, int width) {
    __shared__ float tr[NLAB * 33];         // pad 33 -> conflict-free row reads
    const int t = threadIdx.x;
    for (int i = t; i < NLAB * NLAB; i += 256) {
        int r = i >> 5, c = i & 31;
        tr[r * 33 + c] = trans[i];
    }
    __syncthreads();

    const int wave = t >> 5;
    const int lane = t & 31;
    const int idx  = blockIdx.x * 8 + wave;
    if (idx >= width) return;               // whole wave exits together

    const int node = node_base + idx;
    const int li = left_idx[pos + idx];
    const int ri = right_idx[pos + idx];
    const float cl = const32[lane];

    float lsv = scores[(size_t)li * NLAB + lane];
    float rsv = scores[(size_t)ri * NLAB + lane];
    if (li < N_LEAVES) lsv += cl;           // leaf children store raw emission
    if (ri < N_LEAVES) rsv += cl;
    const float* trow = &tr[lane * 33];

    float lmax = -INFINITY, rmax = -INFINITY;
    #pragma unroll
    for (int k = 0; k < NLAB; ++k) {
        const float tk = trow[k];
        lmax = fmaxf(lmax, tk + __shfl(lsv, k, 32));
        rmax = fmaxf(rmax, tk + __shfl(rsv, k, 32));
    }
    float lsum = 0.0f, rsum = 0.0f;
    #pragma unroll
    for (int k = 0; k < NLAB; ++k) {
        const float tk = trow[k];
        lsum += __expf(tk + __shfl(lsv, k, 32) - lmax);
        rsum += __expf(tk + __shfl(rsv, k, 32) - rmax);
    }
    const float left_m  = __logf(lsum) + lmax;
    const float right_m = __logf(rsum) + rmax;

    scores[(size_t)node * NLAB + lane] += cl + left_m + right_m;
}

// ---------------------------------------------------------------------------
// Launcher
// ---------------------------------------------------------------------------
extern "C" void kernel_launch(void* const* d_in, const int* in_sizes, int n_in,
                              void* d_out, int out_size, void* d_ws, size_t ws_size,
                              hipStream_t stream) {
    const float* hidden    = (const float*)d_in[0];
    const float* trans     = (const float*)d_in[1];
    const float* W_pred    = (const float*)d_in[2];
    const float* b_pred    = (const float*)d_in[3];
    const int*   left_idx  = (const int*)d_in[4];
    const int*   right_idx = (const int*)d_in[5];

    float* scores  = (float*)d_ws;                           // N_NODES*32 floats
    float* partial = scores + (size_t)N_NODES * NLAB;        // 512*1024 floats
    float* colsum  = partial + (size_t)512 * DIM;            // 1024 floats
    float* const32 = colsum + DIM;                           // 32 floats

    // 2048 row tiles of 16 / 4 waves per block -> 512 blocks
    emission_wmma_kernel<<<512, 128, 0, stream>>>(hidden, W_pred, scores, partial);
    colsum_reduce_kernel<<<32, 1024, 0, stream>>>(partial, hidden, colsum);
    const_vec_kernel<<<1, 1024, 0, stream>>>(colsum, W_pred, b_pred, const32);

    int pos = 0, base = N_LEAVES;
    for (int width = N_LEAVES / 2; width >= 1; width >>= 1) {
        int blocks = (width + 7) / 8;
        level_kernel<<<blocks, 256, 0, stream>>>(scores, trans, const32,
                                                 left_idx, right_idx,
                                                 pos, base, width);
        pos  += width;
        base += width;
    }

    (void)hipMemcpyAsync(d_out, scores + (size_t)(N_NODES - 1) * NLAB,
                         NLAB * sizeof(float), hipMemcpyDeviceToDevice, stream);
}